// BatchBeliefStep_7765300871584
// MI455X (gfx1250) — compile-verified
//
#include <hip/hip_runtime.h>
#include <math.h>

#define DT_F    0.1f
#define EPS_F   1e-6f
#define PI_F    3.14159265358979323846f
#define TWOPI_F 6.28318530717958647692f

#ifndef __has_builtin
#define __has_builtin(x) 0
#endif

#if __has_builtin(__builtin_amdgcn_global_load_async_to_lds_b128) && \
    __has_builtin(__builtin_amdgcn_global_store_async_from_lds_b128) && \
    __has_builtin(__builtin_amdgcn_s_wait_asynccnt)
#define HAVE_ASYNC_LDS 1
#else
#define HAVE_ASYNC_LDS 0
#endif

#if __has_builtin(__builtin_amdgcn_tensor_load_to_lds) && \
    __has_builtin(__builtin_amdgcn_tensor_store_from_lds) && \
    __has_builtin(__builtin_amdgcn_s_wait_tensorcnt)
#define HAVE_TDM 1
#else
#define HAVE_TDM 0
#endif

typedef int v4i __attribute__((vector_size(16)));
typedef __attribute__((address_space(1))) v4i GV4;   // global-AS 16B vector
typedef __attribute__((address_space(3))) v4i LV4;   // LDS-AS    16B vector
typedef __attribute__((address_space(3))) const float LCF;

__device__ __forceinline__ void copy16_g2l(const float* g, float* l) {
#if HAVE_ASYNC_LDS
    __builtin_amdgcn_global_load_async_to_lds_b128((GV4*)g, (LV4*)l, 0, 0);
#else
    *(float4*)l = *(const float4*)g;
#endif
}

__device__ __forceinline__ void copy16_l2g(float* g, const float* l) {
#if HAVE_ASYNC_LDS
    __builtin_amdgcn_global_store_async_from_lds_b128((GV4*)g, (LV4*)l, 0, 0);
#else
    *(float4*)g = *(const float4*)l;
#endif
}

__device__ __forceinline__ void wait_async_all() {
#if HAVE_ASYNC_LDS
    __builtin_amdgcn_s_wait_asynccnt(0);
#endif
}

#if HAVE_TDM
typedef unsigned int tdm_v4u __attribute__((ext_vector_type(4)));
typedef int          tdm_v8i __attribute__((ext_vector_type(8)));
typedef int          tdm_v4i __attribute__((ext_vector_type(4)));

// 1-D TDM tile copy: `dwords` DWORDs between global memory and LDS.
// D# per CDNA5 ISA ch.8: group0 = count/lds_addr/global_addr/type,
// group1 = data_size + dims/strides, groups 2/3 unused (<=2D tensor).
// This toolchain exposes the 6-arg builtin: (g0, g1, g2, g3, g4, cpol).
__device__ __forceinline__ void tdm_copy_1d(const void* gaddr, unsigned lds_byte_off,
                                            unsigned dwords, bool is_load) {
    unsigned long long ga = (unsigned long long)(uintptr_t)gaddr;
    tdm_v4u g0;
    g0.x = 1u;                                           // count=1, user mode
    g0.y = lds_byte_off;                                 // lds_addr [63:32]
    g0.z = (unsigned)(ga & 0xFFFFFFFFu);                 // global_addr lo
    g0.w = (unsigned)((ga >> 32) & 0x01FFFFFFu)          // global_addr [56:32]
         | (2u << 30);                                   // type=2 ("image")
    tdm_v8i g1;
    g1[0] = (int)(2u << 16);                             // wg_mask=0, data_size=2 (4B)
    g1[1] = (int)((dwords & 0xFFFFu) << 16);             // tensor_dim0 lo16 @ bits63:48
    g1[2] = (int)(((dwords >> 16) & 0xFFFFu)             // tensor_dim0 hi16
         | (1u << 16));                                  // tensor_dim1 = 1
    g1[3] = (int)((dwords & 0xFFFFu) << 16);             // tile_dim0 @ bits127:112
    g1[4] = 1;                                           // tile_dim1 = 1, tile_dim2 = 0
    g1[5] = (int)dwords;                                 // tensor_dim0_stride lo32
    g1[6] = 0;                                           // stride0 hi16, stride1 lo16
    g1[7] = 0;
    tdm_v4i gz4 = {0, 0, 0, 0};
    tdm_v8i gz8 = {0, 0, 0, 0, 0, 0, 0, 0};
    if (is_load) __builtin_amdgcn_tensor_load_to_lds(g0, g1, gz4, gz4, gz8, 0);
    else         __builtin_amdgcn_tensor_store_from_lds(g0, g1, gz4, gz4, gz8, 0);
}

__device__ __forceinline__ void wait_tensor_all() {
    __builtin_amdgcn_s_wait_tensorcnt(0);
}
#endif

__device__ __forceinline__ unsigned lds_byte_offset(const float* p) {
    return (unsigned)(uintptr_t)(LCF*)p;                 // AS(3) ptr -> 32-bit LDS offset
}

__device__ __forceinline__ float range_angle(float a) {
    float m = fmodf(a + PI_F, TWOPI_F);
    m = (m < 0.0f) ? (m + TWOPI_F) : m;   // python-style mod: result in [0, 2pi)
    return m - PI_F;
}

__global__ __launch_bounds__(256) void ekf_step_kernel(
    const float* __restrict__ x,  const float* __restrict__ P,
    const float* __restrict__ a,  const float* __restrict__ Y,
    const float* __restrict__ n1, const float* __restrict__ n2s,
    const float* __restrict__ n2k,const float* __restrict__ gains,
    const float* __restrict__ og, float* __restrict__ out, int N)
{
    const int tid = threadIdx.x;
    const int blk = blockIdx.x;
    const int n   = blk * 256 + tid;
    const int fullBlocks = N >> 8;                 // N / 256
    const bool fullTile  = (blk < fullBlocks);

    float* __restrict__ xOut = out;
    float* __restrict__ POut = out + (size_t)5  * N;
    float* __restrict__ KOut = out + (size_t)30 * N;

    // Tiles: 256 elements per workgroup.
    __shared__ __align__(16) float sP[256 * 25];   // 25600 B (in, reused for P_new)
    __shared__ __align__(16) float sX[256 * 5];    //  5120 B (in, reused for x_new)
    __shared__ __align__(16) float sA[256 * 2];    //  2048 B
    __shared__ __align__(16) float sK[256 * 10];   // 10240 B

    float x0, x1, x2, a0, a1;
    float p00, p01, p02, p10, p11, p12, p20, p21, p22;

    if (fullTile) {
        const float* Pg = P + (size_t)blk * 6400;  // 25600 B tile
        const float* Xg = x + (size_t)blk * 1280;  //  320 x 16B
        const float* Ag = a + (size_t)blk * 512;   //  128 x 16B
#if HAVE_TDM
        // One DMA descriptor moves the whole P tile; wave 0 issues & tracks it.
        if (tid < 32) tdm_copy_1d(Pg, lds_byte_offset(sP), 6400, true);
#else
        #pragma unroll
        for (int i = tid; i < 1600; i += 256) copy16_g2l(Pg + 4*i, sP + 4*i);
#endif
        for (int i = tid; i < 320;  i += 256) copy16_g2l(Xg + 4*i, sX + 4*i);
        if (tid < 128)                        copy16_g2l(Ag + 4*tid, sA + 4*tid);
        wait_async_all();
#if HAVE_TDM
        if (tid < 32) wait_tensor_all();
#endif
        __syncthreads();

        const float* mx = sX + tid * 5;
        x0 = mx[0]; x1 = mx[1]; x2 = mx[2];
        a0 = sA[tid*2 + 0]; a1 = sA[tid*2 + 1];
        const float* mp = sP + tid * 25;           // stride 25 dwords: bank-conflict free
        p00 = mp[0];  p01 = mp[1];  p02 = mp[2];
        p10 = mp[5];  p11 = mp[6];  p12 = mp[7];
        p20 = mp[10]; p21 = mp[11]; p22 = mp[12];
    } else {
        if (n >= N) return;                        // only tail block diverges; no barriers here
        const float* mx = x + (size_t)n * 5;
        x0 = mx[0]; x1 = mx[1]; x2 = mx[2];
        a0 = a[(size_t)n*2 + 0]; a1 = a[(size_t)n*2 + 1];
        const float* mp = P + (size_t)n * 25;
        p00 = mp[0];  p01 = mp[1];  p02 = mp[2];
        p10 = mp[5];  p11 = mp[6];  p12 = mp[7];
        p20 = mp[10]; p21 = mp[11]; p22 = mp[12];
    }

    // Uniform (wave-invariant) parameters -> scalar loads.
    const float g0 = gains[0], g1 = gains[1];
    const float og0 = og[0],   og1 = og[1];
    const float q0 = expf(2.0f * n1[0]);
    const float q1 = expf(2.0f * n1[1]);
    const float q2 = expf(2.0f * n1[2]);
    const float q3 = expf(2.0f * n1[3]);
    const float q4 = expf(2.0f * n1[4]);
    const float es0 = expf(2.0f * n2s[0]), es1 = expf(2.0f * n2s[1]);
    const float ek0 = expf(2.0f * n2k[0]), ek1 = expf(2.0f * n2k[1]);

    // --- state propagation ---
    const float vel    = g0 * a0;
    const float angvel = g1 * a1;
    const float ang    = range_angle(x2 + angvel * DT_F);
    float sn, cs; sincosf(ang, &sn, &cs);
    const float px = fminf(fmaxf(x0 + vel * cs * DT_F, -1.0f), 1.0f);
    const float py = fminf(fmaxf(x1 + vel * sn * DT_F, -1.0f), 1.0f);
    const float u0 = -vel * sn * DT_F;             // A[0,2]
    const float u1 =  vel * cs * DT_F;             // A[1,2]

    // --- covariance propagation: rows/cols 3,4 of A are zero, so only the
    // symmetrized 3x3 block matters: Ms = A3 * sym(P3) * A3^T, A3 = I + [u0,u1,0] e2^T
    const float s01 = 0.5f * (p01 + p10);
    const float s02 = 0.5f * (p02 + p20);
    const float s12 = 0.5f * (p12 + p21);
    const float m00 = p00 + 2.0f*u0*s02 + u0*u0*p22;
    const float m01 = s01 + u0*s12 + s02*u1 + u0*u1*p22;
    const float m02 = s02 + u0*p22;
    const float m11 = p11 + 2.0f*u1*s12 + u1*u1*p22;
    const float m12 = s12 + u1*p22;
    const float m22 = p22;

    // --- S is exactly diagonal; K has exactly two nonzeros ---
    const float S00 = q3*og0*og0 + vel*vel*ek0 + es0;
    const float S11 = q4*og1*og1 + angvel*angvel*ek1 + es1;
    const float K30 = (q3 * og0) / S00;
    const float K41 = (q4 * og1) / S11;

    const float y0 = Y[n];           // Y is (2,N): perfectly coalesced
    const float y1 = Y[(size_t)N + n];
    const float e0 = y0 - og0 * vel;
    const float e1 = y1 - og1 * angvel;
    const float x3n = vel    + K30 * e0;
    const float x4n = angvel + K41 * e1;
    const float P33 = (1.0f - K30 * og0) * q3 + EPS_F;
    const float P44 = (1.0f - K41 * og1) * q4 + EPS_F;

    const float d00 = m00 + q0 + EPS_F;
    const float d11 = m11 + q1 + EPS_F;
    const float d22 = m22 + q2 + EPS_F;

    if (fullTile) {
        // Write results into LDS tiles, then stream back (TDM for P, async for x/K).
        float* ox = sX + tid * 5;
        ox[0] = px; ox[1] = py; ox[2] = ang; ox[3] = x3n; ox[4] = x4n;
        float* op = sP + tid * 25;
        op[0]=d00; op[1]=m01; op[2]=m02; op[3]=0.0f; op[4]=0.0f;
        op[5]=m01; op[6]=d11; op[7]=m12; op[8]=0.0f; op[9]=0.0f;
        op[10]=m02; op[11]=m12; op[12]=d22; op[13]=0.0f; op[14]=0.0f;
        op[15]=0.0f; op[16]=0.0f; op[17]=0.0f; op[18]=P33; op[19]=0.0f;
        op[20]=0.0f; op[21]=0.0f; op[22]=0.0f; op[23]=0.0f; op[24]=P44;
        float* ok = sK + tid * 10;
        #pragma unroll
        for (int i = 0; i < 10; ++i) ok[i] = 0.0f;
        ok[6] = K30;   // K[3,0]
        ok[9] = K41;   // K[4,1]
        __syncthreads();

        float* Pg = POut + (size_t)blk * 6400;
        float* Xg = xOut + (size_t)blk * 1280;
        float* Kg = KOut + (size_t)blk * 2560;
#if HAVE_TDM
        if (tid < 32) tdm_copy_1d(Pg, lds_byte_offset(sP), 6400, false);
#else
        for (int i = tid; i < 1600; i += 256) copy16_l2g(Pg + 4*i, sP + 4*i);
#endif
        for (int i = tid; i < 320;  i += 256) copy16_l2g(Xg + 4*i, sX + 4*i);
        for (int i = tid; i < 640;  i += 256) copy16_l2g(Kg + 4*i, sK + 4*i);
        wait_async_all();   // S_ENDPGM also implies wait-idle, but be explicit
#if HAVE_TDM
        if (tid < 32) wait_tensor_all();
#endif
    } else {
        float* ox = xOut + (size_t)n * 5;
        ox[0] = px; ox[1] = py; ox[2] = ang; ox[3] = x3n; ox[4] = x4n;
        float* op = POut + (size_t)n * 25;
        op[0]=d00; op[1]=m01; op[2]=m02; op[3]=0.0f; op[4]=0.0f;
        op[5]=m01; op[6]=d11; op[7]=m12; op[8]=0.0f; op[9]=0.0f;
        op[10]=m02; op[11]=m12; op[12]=d22; op[13]=0.0f; op[14]=0.0f;
        op[15]=0.0f; op[16]=0.0f; op[17]=0.0f; op[18]=P33; op[19]=0.0f;
        op[20]=0.0f; op[21]=0.0f; op[22]=0.0f; op[23]=0.0f; op[24]=P44;
        float* ok = KOut + (size_t)n * 10;
        #pragma unroll
        for (int i = 0; i < 10; ++i) ok[i] = 0.0f;
        ok[6] = K30;
        ok[9] = K41;
    }
}

extern "C" void kernel_launch(void* const* d_in, const int* in_sizes, int n_in,
                              void* d_out, int out_size, void* d_ws, size_t ws_size,
                              hipStream_t stream) {
    const float* x    = (const float*)d_in[0];
    const float* P    = (const float*)d_in[1];
    const float* a    = (const float*)d_in[2];
    const float* Y    = (const float*)d_in[3];
    const float* n1   = (const float*)d_in[4];
    const float* n2s  = (const float*)d_in[5];
    const float* n2k  = (const float*)d_in[6];
    const float* g    = (const float*)d_in[7];
    const float* og   = (const float*)d_in[8];
    const int N = in_sizes[0] / 5;
    const int blocks = (N + 255) / 256;
    ekf_step_kernel<<<blocks, 256, 0, stream>>>(x, P, a, Y, n1, n2s, n2k, g, og,
                                                (float*)d_out, N);
}